// PYG_DeepGCNEncoder_55070070670014
// MI455X (gfx1250) — compile-verified
//
#include <hip/hip_runtime.h>
#include <hip/hip_bf16.h>

#define N_NODES 50000
#define N_EDGES 600000
#define DIM 128
#define HID 256
#define NLAYERS 2
#define EPS_MSG 1e-7f
#define EPS_NORM 1e-5f

typedef float v2f __attribute__((ext_vector_type(2)));
typedef float v8f __attribute__((ext_vector_type(8)));

// monotone float<->uint encoding so atomicMax(uint) == float max (any sign)
__device__ __forceinline__ unsigned int enc_f32(float f) {
  unsigned int u = __float_as_uint(f);
  return (u & 0x80000000u) ? ~u : (u | 0x80000000u);
}
__device__ __forceinline__ float dec_f32(unsigned int u) {
  unsigned int b = (u & 0x80000000u) ? (u & 0x7FFFFFFFu) : ~u;
  return __uint_as_float(b);
}

// stats layout (floats): [0,128)=sum  [128,256)=sumsq  [256,384)=scale  [384,512)=shift
__global__ void k_encode(const float* __restrict__ x, const float* __restrict__ Wn,
                         const float* __restrict__ bnb, float* __restrict__ h) {
  int gid = blockIdx.x * 256 + threadIdx.x;
  if (gid >= N_NODES * DIM) return;
  int n = gid >> 7, d = gid & 127;
  h[gid] = x[n] * Wn[d] + bnb[d];
}

__global__ void k_init_layer(unsigned int* __restrict__ m_enc, float* __restrict__ den,
                             float* __restrict__ num, float* __restrict__ stats) {
  int gid = blockIdx.x * 256 + threadIdx.x;
  if (gid < 512) stats[gid] = 0.f;
  if (gid >= N_NODES * DIM) return;
  m_enc[gid] = 0u;   // decodes below any real logit
  den[gid] = 0.f;
  num[gid] = 0.f;
}

__global__ void k_bn_stats(const float* __restrict__ h, float* __restrict__ stats) {
  __shared__ float red[256];
  int tid = threadIdx.x;
  int col = tid & 127;
  int r0 = blockIdx.x * 64 + (tid >> 7);
  float s = 0.f, s2 = 0.f;
  for (int i = 0; i < 32; i++) {
    int r = r0 + 2 * i;
    if (r < N_NODES) { float v = h[r * DIM + col]; s += v; s2 += v * v; }
  }
  red[tid] = s;
  __syncthreads();
  if (tid < 128) atomicAdd(&stats[col], red[tid] + red[tid + 128]);
  __syncthreads();
  red[tid] = s2;
  __syncthreads();
  if (tid < 128) atomicAdd(&stats[128 + col], red[tid] + red[tid + 128]);
}

__global__ void k_bn_finalize(const float* __restrict__ gamma, const float* __restrict__ beta,
                              float* __restrict__ stats) {
  int d = threadIdx.x;
  if (d >= DIM) return;
  float mean = stats[d] * (1.f / (float)N_NODES);
  float var  = stats[128 + d] * (1.f / (float)N_NODES) - mean * mean;
  float rstd = rsqrtf(var + EPS_NORM);
  float sc = gamma[d] * rstd;
  stats[256 + d] = sc;
  stats[384 + d] = beta[d] - mean * sc;
}

__device__ __forceinline__ float edge_msg(const float* __restrict__ h,
                                          const float* __restrict__ stats,
                                          const float* __restrict__ We,
                                          const float* __restrict__ be,
                                          int src, int d, float ea) {
  float hn = fmaxf(h[src * DIM + d] * stats[256 + d] + stats[384 + d], 0.f);
  float e = ea * We[d] + be[d];
  return fmaxf(hn + e, 0.f) + EPS_MSG;
}

__global__ void k_msg_max(const float* __restrict__ h, const int* __restrict__ eidx,
                          const float* __restrict__ eattr, const float* __restrict__ We,
                          const float* __restrict__ be, const float* __restrict__ stats,
                          const float* __restrict__ t, unsigned int* __restrict__ m_enc) {
  int gid = blockIdx.x * 256 + threadIdx.x;
  if (gid >= N_EDGES * DIM) return;
  int e = gid >> 7, d = gid & 127;
  int src = eidx[e], dst = eidx[N_EDGES + e];
  float msg = edge_msg(h, stats, We, be, src, d, eattr[e]);
  atomicMax(&m_enc[dst * DIM + d], enc_f32(msg * t[0]));
}

__global__ void k_msg_sum(const float* __restrict__ h, const int* __restrict__ eidx,
                          const float* __restrict__ eattr, const float* __restrict__ We,
                          const float* __restrict__ be, const float* __restrict__ stats,
                          const float* __restrict__ t, const unsigned int* __restrict__ m_enc,
                          float* __restrict__ den, float* __restrict__ num) {
  int gid = blockIdx.x * 256 + threadIdx.x;
  if (gid >= N_EDGES * DIM) return;
  int e = gid >> 7, d = gid & 127;
  int src = eidx[e], dst = eidx[N_EDGES + e];
  float msg = edge_msg(h, stats, We, be, src, d, eattr[e]);
  float m = dec_f32(m_enc[dst * DIM + d]);
  float ex = expf(msg * t[0] - m);
  atomicAdd(&den[dst * DIM + d], ex);
  atomicAdd(&num[dst * DIM + d], msg * ex);
}

// One block = 16 rows.  GEMM1 (16x128 @ 128x256) -> LN+ReLU -> GEMM2 (16x256 @ 256x128),
// residual h += z.  4 wave32 per block; fp32 WMMA 16x16x4.
__global__ __launch_bounds__(128)
void k_mlp(float* __restrict__ h, const float* __restrict__ den, const float* __restrict__ num,
           const float* __restrict__ stats,
           const float* __restrict__ W1, const float* __restrict__ b1,
           const float* __restrict__ lng, const float* __restrict__ lnb,
           const float* __restrict__ W2, const float* __restrict__ b2) {
  __shared__ float A_s[16 * DIM];
  __shared__ float Z_s[16 * HID];
  __shared__ float lr1[16][8], lr2[16][8];
  __shared__ float mu_s[16], rs_s[16];

  int tid = threadIdx.x;
  int row0 = blockIdx.x * 16;

  // A = softmax-aggregate + hn   (fused: never materialized in HBM)
  for (int i = 0; i < 16; i++) {
    int idx = tid + i * 128;
    int m = idx >> 7, d = idx & 127;
    int g = (row0 + m) * DIM + d;
    float agg = num[g] / (den[g] + 1e-16f);
    float hn = fmaxf(h[g] * stats[256 + d] + stats[384 + d], 0.f);
    A_s[idx] = agg + hn;
  }
  __syncthreads();

  int w = tid >> 5, lane = tid & 31;
  int hi = lane >> 4, l16 = lane & 15;

  // GEMM1: each wave -> 4 column tiles of 16; K=128 in steps of 4
  for (int nt = w * 4; nt < w * 4 + 4; nt++) {
    int n = nt * 16 + l16;
    v8f acc = {0.f, 0.f, 0.f, 0.f, 0.f, 0.f, 0.f, 0.f};
    for (int k4 = 0; k4 < DIM; k4 += 4) {
      int ka = k4 + 2 * hi;
      v2f a = *(const v2f*)&A_s[l16 * DIM + ka];       // A[m][ka], A[m][ka+1]
      v2f b;
      b.x = W1[ka * HID + n];                           // B[ka][n]
      b.y = W1[(ka + 1) * HID + n];                     // B[ka+1][n]
      acc = __builtin_amdgcn_wmma_f32_16x16x4_f32(false, a, false, b, (short)0, acc,
                                                  false, false);
    }
    float bias = b1[n];
    for (int r = 0; r < 8; r++) Z_s[(r + hi * 8) * HID + n] = acc[r] + bias;
  }
  __syncthreads();

  // LayerNorm(256) + ReLU in LDS
  {
    int row = tid >> 3, sub = tid & 7;
    float s = 0.f, s2 = 0.f;
    for (int c = sub * 32; c < sub * 32 + 32; c++) {
      float v = Z_s[row * HID + c]; s += v; s2 += v * v;
    }
    lr1[row][sub] = s; lr2[row][sub] = s2;
    __syncthreads();
    if (tid < 16) {
      float ts = 0.f, ts2 = 0.f;
      for (int j = 0; j < 8; j++) { ts += lr1[tid][j]; ts2 += lr2[tid][j]; }
      float mean = ts * (1.f / (float)HID);
      float var = ts2 * (1.f / (float)HID) - mean * mean;
      mu_s[tid] = mean;
      rs_s[tid] = rsqrtf(var + EPS_NORM);
    }
    __syncthreads();
    for (int i = 0; i < 32; i++) {
      int idx = tid + i * 128;
      int row2 = idx >> 8, col = idx & 255;
      float z = (Z_s[idx] - mu_s[row2]) * rs_s[row2] * lng[col] + lnb[col];
      Z_s[idx] = fmaxf(z, 0.f);
    }
  }
  __syncthreads();

  // GEMM2: each wave -> 2 column tiles of 16; K=256; fused residual h += z + b2
  for (int nt = w * 2; nt < w * 2 + 2; nt++) {
    int n = nt * 16 + l16;
    v8f acc = {0.f, 0.f, 0.f, 0.f, 0.f, 0.f, 0.f, 0.f};
    for (int k4 = 0; k4 < HID; k4 += 4) {
      int ka = k4 + 2 * hi;
      v2f a = *(const v2f*)&Z_s[l16 * HID + ka];
      v2f b;
      b.x = W2[ka * DIM + n];
      b.y = W2[(ka + 1) * DIM + n];
      acc = __builtin_amdgcn_wmma_f32_16x16x4_f32(false, a, false, b, (short)0, acc,
                                                  false, false);
    }
    float bias = b2[n];
    for (int r = 0; r < 8; r++) {
      int g = (row0 + r + hi * 8) * DIM + n;
      h[g] += acc[r] + bias;
    }
  }
}

__global__ void k_edge_dot(const float* __restrict__ h, const int* __restrict__ eidx,
                           float* __restrict__ out) {
  int tid = threadIdx.x;
  int e = blockIdx.x * 8 + (tid >> 5);  // one wave32 per edge
  int lane = tid & 31;
  if (e >= N_EDGES) return;
  int src = eidx[e], dst = eidx[N_EDGES + e];
  float s = 0.f;
  for (int i = 0; i < 4; i++) {
    int c = lane + i * 32;
    s += h[src * DIM + c] * h[dst * DIM + c];
  }
  for (int off = 16; off; off >>= 1) s += __shfl_down(s, off, 32);
  if (lane == 0) out[e] = s;
}

extern "C" void kernel_launch(void* const* d_in, const int* in_sizes, int n_in,
                              void* d_out, int out_size, void* d_ws, size_t ws_size,
                              hipStream_t stream) {
  const float* x      = (const float*)d_in[0];
  const int*   eidx   = (const int*)d_in[1];
  const float* eattr  = (const float*)d_in[2];
  const float* Wn     = (const float*)d_in[3];
  const float* bnb    = (const float*)d_in[4];
  const float* We     = (const float*)d_in[5];
  const float* be     = (const float*)d_in[6];
  const float* bng    = (const float*)d_in[7];
  const float* bnbeta = (const float*)d_in[8];
  const float* t      = (const float*)d_in[9];
  const float* W1     = (const float*)d_in[10];
  const float* b1     = (const float*)d_in[11];
  const float* lng    = (const float*)d_in[12];
  const float* lnb    = (const float*)d_in[13];
  const float* W2     = (const float*)d_in[14];
  const float* b2     = (const float*)d_in[15];
  float* out = (float*)d_out;

  const size_t ND = (size_t)N_NODES * DIM;
  float*        h     = (float*)d_ws;
  unsigned int* m_enc = (unsigned int*)(h + ND);
  float*        den   = (float*)(m_enc + ND);
  float*        num   = den + ND;
  float*        stats = num + ND;   // 512 floats

  const int nd_blocks = (int)((ND + 255) / 256);                 // 25000
  const int ed_blocks = (int)(((size_t)N_EDGES * DIM + 255) / 256); // 300000

  k_encode<<<nd_blocks, 256, 0, stream>>>(x, Wn, bnb, h);

  for (int l = 0; l < NLAYERS; l++) {
    k_init_layer<<<nd_blocks, 256, 0, stream>>>(m_enc, den, num, stats);
    k_bn_stats<<<(N_NODES + 63) / 64, 256, 0, stream>>>(h, stats);
    k_bn_finalize<<<1, 128, 0, stream>>>(bng + l * DIM, bnbeta + l * DIM, stats);
    k_msg_max<<<ed_blocks, 256, 0, stream>>>(h, eidx, eattr, We, be, stats, t + l, m_enc);
    k_msg_sum<<<ed_blocks, 256, 0, stream>>>(h, eidx, eattr, We, be, stats, t + l, m_enc,
                                             den, num);
    k_mlp<<<N_NODES / 16, 128, 0, stream>>>(h, den, num, stats,
        W1 + (size_t)l * DIM * HID, b1 + l * HID, lng + l * HID, lnb + l * HID,
        W2 + (size_t)l * HID * DIM, b2 + l * DIM);
  }

  k_edge_dot<<<(N_EDGES + 7) / 8, 256, 0, stream>>>(h, eidx, out);
}